// GraphMem_48455821033613
// MI455X (gfx1250) — compile-verified
//
#include <hip/hip_runtime.h>
#include <hip/hip_bf16.h>
#include <math.h>

// ---- problem constants (match reference) ----
#define Bq   8
#define Eq   2000
#define Tq   10
#define Dq   200
#define WDq  300
#define KDq  100
#define NRP1 201      // NR+1 relations
#define NEq  200000
#define Lq   3
#define VNEGf (-100000000000.0f)

typedef float v2f __attribute__((ext_vector_type(2)));
typedef float v8f __attribute__((ext_vector_type(8)));

static __device__ __forceinline__ float sigmoidf_(float x) {
  return 1.0f / (1.0f + expf(-x));
}

// =====================================================================
// 1) BiLSTM query encoder -> q_node [B, D]. One block per batch.
// =====================================================================
__global__ void k_lstm(const int* __restrict__ qtext, const float* __restrict__ wemb,
                       const float* __restrict__ Wih_f, const float* __restrict__ Whh_f,
                       const float* __restrict__ bih_f, const float* __restrict__ bhh_f,
                       const float* __restrict__ Wih_b, const float* __restrict__ Whh_b,
                       const float* __restrict__ bih_b, const float* __restrict__ bhh_b,
                       float* __restrict__ qnode) {
  __shared__ float x[WDq];
  __shared__ float h[Dq], c[Dq], g[4 * Dq], hf[Dq];
  const int b = blockIdx.x, tid = threadIdx.x;
  for (int dir = 0; dir < 2; ++dir) {
    const float* Wih = dir ? Wih_b : Wih_f;
    const float* Whh = dir ? Whh_b : Whh_f;
    const float* bih = dir ? bih_b : bih_f;
    const float* bhh = dir ? bhh_b : bhh_f;
    for (int d = tid; d < Dq; d += blockDim.x) { h[d] = 0.f; c[d] = 0.f; }
    __syncthreads();
    for (int t = 0; t < Tq; ++t) {
      const int tt = dir ? (Tq - 1 - t) : t;
      const int w = qtext[b * Tq + tt];
      const float* xr = wemb + (size_t)w * WDq;
      for (int i = tid; i < WDq; i += blockDim.x) x[i] = xr[i];
      __syncthreads();
      for (int gi = tid; gi < 4 * Dq; gi += blockDim.x) {
        const float* wi = Wih + (size_t)gi * WDq;
        const float* wh = Whh + (size_t)gi * Dq;
        float acc = bih[gi] + bhh[gi];
        for (int k = 0; k < WDq; ++k) acc += wi[k] * x[k];
        for (int k = 0; k < Dq; ++k)  acc += wh[k] * h[k];
        g[gi] = acc;
      }
      __syncthreads();
      for (int d = tid; d < Dq; d += blockDim.x) {
        const float ig = sigmoidf_(g[d]);
        const float fg = sigmoidf_(g[Dq + d]);
        const float gg = tanhf(g[2 * Dq + d]);
        const float og = sigmoidf_(g[3 * Dq + d]);
        const float cn = fg * c[d] + ig * gg;
        c[d] = cn;
        h[d] = og * tanhf(cn);
      }
      __syncthreads();
    }
    if (dir == 0) {
      for (int d = tid; d < Dq; d += blockDim.x) hf[d] = h[d];
      __syncthreads();
    } else {
      for (int d = tid; d < Dq; d += blockDim.x)
        qnode[b * Dq + d] = 0.5f * (hf[d] + h[d]);
    }
  }
}

// =====================================================================
// 2) rel_proj = rel_emb @ W_rel^T + b_rel  via V_WMMA_F32_16X16X4_F32
//    M=201 (pad to 208), N=200 (pad to 208), K=300. One wave per tile.
//    OOB rows: pointers clamped + stores land in the 208-row padded ws
//    region (never read back), so loads AND stores are unconditional.
// =====================================================================
__global__ void k_relproj(const float* __restrict__ relemb, const float* __restrict__ Wrel,
                          const float* __restrict__ brel, float* __restrict__ relproj) {
  const int wave = (blockIdx.x * blockDim.x + threadIdx.x) >> 5;
  const int lane = threadIdx.x & 31;
  const int MT = 13, NT = 13;
  if (wave >= MT * NT) return;            // whole-wave exit, EXEC stays all-1s
  const int mt = wave / NT, nt = wave % NT;
  const int half = lane >> 4, lrow = lane & 15;
  const int m = mt * 16 + lrow;           // A row (lanes 16-31 carry same M, K+2..K+3)
  const int n = nt * 16 + lrow;           // B col
  const bool mval = (m < NRP1);
  const bool nval = (n < Dq);
  const v2f* pa = (const v2f*)(relemb + (size_t)(mval ? m : 0) * WDq);
  const v2f* pb = (const v2f*)(Wrel   + (size_t)(nval ? n : 0) * WDq);
  v8f acc = {};
#pragma unroll 5
  for (int k = 0; k < WDq; k += 4) {
    const v2f a  = pa[(k >> 1) + half];   // A[m][k+2h], A[m][k+2h+1]
    const v2f bb = pb[(k >> 1) + half];   // B[k+2h][n] = Wrel[n][k+2h], +1
    acc = __builtin_amdgcn_wmma_f32_16x16x4_f32(false, a, false, bb, (short)0, acc,
                                                false, false);
  }
  if (nval) {                             // single lane predicate for all 8 stores
    const float bias = brel[n];
#pragma unroll
    for (int r = 0; r < 8; ++r) {
      const int row = mt * 16 + r + 8 * half;   // C/D: lanes16-31 hold M+8
      relproj[row * Dq + n] = acc[r] + bias;    // rows 201..207 hit padded ws
    }
  }
}

// =====================================================================
// 3) ent_emb = entity_kge[local_entity] @ W_ent^T + b_ent
//    M = B*E = 16000 (exactly 1000 tiles), K = 100, N = 200 (13 tiles).
// =====================================================================
__global__ void k_entemb(const int* __restrict__ local_entity, const float* __restrict__ kge,
                         const float* __restrict__ Went, const float* __restrict__ bent,
                         float* __restrict__ entemb) {
  const int wave = (blockIdx.x * blockDim.x + threadIdx.x) >> 5;
  const int lane = threadIdx.x & 31;
  const int NT = 13;
  const int MTOT = (Bq * Eq) / 16;   // 1000
  if (wave >= MTOT * NT) return;
  const int mt = wave / NT, nt = wave % NT;
  const int half = lane >> 4, lrow = lane & 15;
  const int m = mt * 16 + lrow;                  // always < 16000
  const int idx = local_entity[m];
  const int n = nt * 16 + lrow;
  const bool nval = (n < Dq);
  const v2f* pa = (const v2f*)(kge  + (size_t)idx * KDq);
  const v2f* pb = (const v2f*)(Went + (size_t)(nval ? n : 0) * KDq);
  v8f acc = {};
#pragma unroll 5
  for (int k = 0; k < KDq; k += 4) {             // 25 WMMA steps
    const v2f a  = pa[(k >> 1) + half];
    const v2f bb = pb[(k >> 1) + half];
    acc = __builtin_amdgcn_wmma_f32_16x16x4_f32(false, a, false, bb, (short)0, acc,
                                                false, false);
  }
  if (nval) {
    const float bias = bent[n];
#pragma unroll
    for (int r = 0; r < 8; ++r) {
      const int row = mt * 16 + r + 8 * half;
      entemb[(size_t)row * Dq + n] = acc[r] + bias;
    }
  }
}

// =====================================================================
// 4) entity_state[b,d] = sum_e start[b,e] * ent_emb[b,e,d]
// =====================================================================
__global__ void k_estate(const float* __restrict__ start, const float* __restrict__ entemb,
                         float* __restrict__ estate) {
  const int b = blockIdx.x, d = threadIdx.x;
  if (d >= Dq) return;
  const float* em = entemb + (size_t)b * Eq * Dq;
  const float* st = start + b * Eq;
  float acc = 0.f;
  for (int e = 0; e < Eq; ++e) {
    const float s = st[e];                 // uniform per-e branch: skip ~99% rows
    if (s != 0.f) acc += s * em[(size_t)e * Dq + d];
  }
  estate[b * Dq + d] = acc;
}

// =====================================================================
// 5) GRU decoder over relation attention; latent[b,d]. One block per b.
// =====================================================================
__global__ void k_gru(const float* __restrict__ qnode, const float* __restrict__ relproj,
                      const float* __restrict__ Wih, const float* __restrict__ Whh,
                      const float* __restrict__ bih, const float* __restrict__ bhh,
                      const float* __restrict__ w_state, const float* __restrict__ w_query,
                      float* __restrict__ latent) {
  __shared__ float hid[Dq], din[Dq], gi[3 * Dq], gh[3 * Dq], hist[Dq];
  __shared__ float attn[208];
  __shared__ float smax, ssum;
  const int b = blockIdx.x, tid = threadIdx.x;
  for (int d = tid; d < Dq; d += blockDim.x) { hid[d] = qnode[b * Dq + d]; din[d] = 0.f; }
  __syncthreads();
  const float inv_sqrt_d = rsqrtf((float)Dq);
  for (int l = 0; l < Lq; ++l) {
    for (int g = tid; g < 3 * Dq; g += blockDim.x) {
      const float* wi = Wih + (size_t)g * Dq;
      const float* wh = Whh + (size_t)g * Dq;
      float ai = bih[g], ah = bhh[g];
      for (int k = 0; k < Dq; ++k) { ai += wi[k] * din[k]; ah += wh[k] * hid[k]; }
      gi[g] = ai; gh[g] = ah;
    }
    __syncthreads();
    for (int d = tid; d < Dq; d += blockDim.x) {
      const float r = sigmoidf_(gi[d] + gh[d]);
      const float z = sigmoidf_(gi[Dq + d] + gh[Dq + d]);
      const float n = tanhf(gi[2 * Dq + d] + r * gh[2 * Dq + d]);
      hid[d] = (1.f - z) * n + z * hid[d];
    }
    __syncthreads();
    if (tid < NRP1) {
      const float* rp = relproj + tid * Dq;
      float a = 0.f;
      for (int k = 0; k < Dq; ++k) a += hid[k] * rp[k];
      attn[tid] = a * inv_sqrt_d;
    }
    __syncthreads();
    if (tid == 0) {
      float mx = attn[0];
      for (int j = 1; j < NRP1; ++j) mx = fmaxf(mx, attn[j]);
      smax = mx;
    }
    __syncthreads();
    if (tid < NRP1) attn[tid] = expf(attn[tid] - smax);
    __syncthreads();
    if (tid == 0) {
      float s = 0.f;
      for (int j = 0; j < NRP1; ++j) s += attn[j];
      ssum = s;
    }
    __syncthreads();
    if (tid < NRP1) attn[tid] /= ssum;
    __syncthreads();
    for (int d = tid; d < Dq; d += blockDim.x) {
      float a = 0.f;
      for (int j = 0; j < NRP1; ++j) a += attn[j] * relproj[j * Dq + d];
      hist[d] = (l == 0) ? a : hist[d] * a;
      din[d] = a;
    }
    __syncthreads();
  }
  const float wst = w_state[0], wq = w_query[0];
  for (int d = tid; d < Dq; d += blockDim.x)
    latent[b * Dq + d] = hist[d] * wst + qnode[b * Dq + d] * wq;
}

// =====================================================================
// 6) scores[b,e] = dot(entity_state[b]*latent[b], ent_emb[b,e,:]); wave/row
// =====================================================================
__global__ void k_scores(const float* __restrict__ estate, const float* __restrict__ latent,
                         const float* __restrict__ entemb, float* __restrict__ scores) {
  const int row = (blockIdx.x * blockDim.x + threadIdx.x) >> 5;
  const int lane = threadIdx.x & 31;
  if (row >= Bq * Eq) return;
  const int b = row / Eq;
  const float* em = entemb + (size_t)row * Dq;
  const float* es = estate + b * Dq;
  const float* la = latent + b * Dq;
  float acc = 0.f;
  for (int d = lane; d < Dq; d += 32) acc += es[d] * la[d] * em[d];
  for (int o = 16; o > 0; o >>= 1) acc += __shfl_down(acc, o, 32);
  if (lane == 0) scores[row] = acc;
}

// =====================================================================
// 7) reachability propagation (3 launches). 8 waves/block = 8 rows of one
//    batch (2000 % 8 == 0); frontier vector staged in LDS; A rows via b128.
// =====================================================================
__global__ void k_prop_init(const float* __restrict__ start, float* __restrict__ s_cur,
                            float* __restrict__ trav) {
  const int i = blockIdx.x * blockDim.x + threadIdx.x;
  if (i < Bq * Eq) { const float v = start[i]; s_cur[i] = v; trav[i] = v; }
}

__global__ void k_prop(const float* __restrict__ e2e, const float* __restrict__ s_in,
                       float* __restrict__ s_out, float* __restrict__ trav) {
  __shared__ float sloc[Eq];
  const int row0 = blockIdx.x * 8;
  const int b = row0 / Eq;                        // whole block in one batch
  for (int i = threadIdx.x; i < Eq; i += blockDim.x) sloc[i] = s_in[b * Eq + i];
  __syncthreads();
  const int wave = threadIdx.x >> 5, lane = threadIdx.x & 31;
  const int row = row0 + wave;
  const float4* arow = (const float4*)(e2e + (size_t)row * Eq);
  const float4* sv   = (const float4*)sloc;
  float acc = 0.f;
  for (int i = lane; i < Eq / 4; i += 32) {       // 500 b128 chunks per row
    const float4 a = arow[i];
    const float4 s = sv[i];
    acc += a.x * s.x + a.y * s.y + a.z * s.z + a.w * s.w;
  }
  for (int o = 16; o > 0; o >>= 1) acc += __shfl_down(acc, o, 32);
  if (lane == 0) {
    const float tv = trav[row];
    const float pre = acc + tv;
    const float endv = ((pre > 0.f) ? 1.f : pre) - tv;
    s_out[row] = endv;
    trav[row] = tv + endv;
  }
}

// =====================================================================
// 8) per-batch: score_m, pred_dist, argmax -> out[1+b], out[9+b*E+e]
// =====================================================================
__global__ void k_final(const float* __restrict__ scores, const int* __restrict__ local_entity,
                        const float* __restrict__ trav, float* __restrict__ out) {
  __shared__ float bmax[256];
  __shared__ int bidx[256];
  const int b = blockIdx.x, tid = threadIdx.x;
  float mx = -INFINITY; int mi = 0;
  for (int e = tid; e < Eq; e += 256) {
    const int r = b * Eq + e;
    const float mask = (local_entity[r] != NEq) ? 1.f : 0.f;
    const float sm = scores[r] + (1.f - mask) * VNEGf + (1.f - trav[r]) * VNEGf;
    out[9 + r] = sigmoidf_(sm) * mask;
    if (sm > mx) { mx = sm; mi = e; }
  }
  bmax[tid] = mx; bidx[tid] = mi;
  __syncthreads();
  for (int s = 128; s > 0; s >>= 1) {
    if (tid < s) {
      if (bmax[tid + s] > bmax[tid] ||
          (bmax[tid + s] == bmax[tid] && bidx[tid + s] < bidx[tid])) {
        bmax[tid] = bmax[tid + s]; bidx[tid] = bidx[tid + s];
      }
    }
    __syncthreads();
  }
  if (tid == 0) out[1 + b] = (float)bidx[0];
}

// =====================================================================
// 9) BCE-with-logits mean -> out[0]. Single block.
// =====================================================================
__global__ void k_loss(const float* __restrict__ scores, const float* __restrict__ y,
                       float* __restrict__ out) {
  __shared__ float red[256];
  const int tid = threadIdx.x;
  float acc = 0.f;
  for (int i = tid; i < Bq * Eq; i += 256) {
    const float x = scores[i];
    acc += fmaxf(x, 0.f) - x * y[i] + log1pf(expf(-fabsf(x)));
  }
  red[tid] = acc; __syncthreads();
  for (int s = 128; s > 0; s >>= 1) { if (tid < s) red[tid] += red[tid + s]; __syncthreads(); }
  if (tid == 0) out[0] = red[0] / (float)(Bq * Eq);
}

// =====================================================================
extern "C" void kernel_launch(void* const* d_in, const int* in_sizes, int n_in,
                              void* d_out, int out_size, void* d_ws, size_t ws_size,
                              hipStream_t stream) {
  (void)in_sizes; (void)n_in; (void)out_size; (void)ws_size;
  const int*   local_entity = (const int*)  d_in[0];
  const float* q2e          = (const float*)d_in[1];   // start [B,E]
  const float* e2e          = (const float*)d_in[2];
  /* d_in[3] kb_fact_rel: unused by reference */
  const int*   query_text   = (const int*)  d_in[4];
  const float* answer_dist  = (const float*)d_in[5];
  const float* word_emb     = (const float*)d_in[6];
  const float* entity_kge   = (const float*)d_in[7];
  const float* W_ent        = (const float*)d_in[8];
  const float* b_ent        = (const float*)d_in[9];
  const float* rel_emb      = (const float*)d_in[10];
  const float* W_rel        = (const float*)d_in[11];
  const float* b_rel        = (const float*)d_in[12];
  const float* W_ih_f       = (const float*)d_in[13];
  const float* W_hh_f       = (const float*)d_in[14];
  const float* b_ih_f       = (const float*)d_in[15];
  const float* b_hh_f       = (const float*)d_in[16];
  const float* W_ih_b       = (const float*)d_in[17];
  const float* W_hh_b       = (const float*)d_in[18];
  const float* b_ih_b       = (const float*)d_in[19];
  const float* b_hh_b       = (const float*)d_in[20];
  const float* W_ih_g       = (const float*)d_in[21];
  const float* W_hh_g       = (const float*)d_in[22];
  const float* b_ih_g       = (const float*)d_in[23];
  const float* b_hh_g       = (const float*)d_in[24];
  const float* w_state      = (const float*)d_in[25];
  const float* w_query      = (const float*)d_in[26];

  float* out = (float*)d_out;
  float* ws  = (float*)d_ws;

  // workspace layout (floats)
  float* qnode   = ws;                         // 1600
  float* relproj = qnode   + 1600;             // 208*200 = 41600 (padded)
  float* entemb  = relproj + 41600;            // 3,200,000
  float* estate  = entemb  + (size_t)Bq * Eq * Dq;  // 1600
  float* latent  = estate  + 1600;             // 1600
  float* scores  = latent  + 1600;             // 16000
  float* s_a     = scores  + Bq * Eq;          // 16000
  float* s_b     = s_a     + Bq * Eq;          // 16000
  float* trav    = s_b     + Bq * Eq;          // 16000

  // 1) query encoder
  k_lstm<<<Bq, 256, 0, stream>>>(query_text, word_emb,
                                 W_ih_f, W_hh_f, b_ih_f, b_hh_f,
                                 W_ih_b, W_hh_b, b_ih_b, b_hh_b, qnode);
  // 2) rel_proj GEMM (169 tiles, 1 wave each)
  k_relproj<<<22, 256, 0, stream>>>(rel_emb, W_rel, b_rel, relproj);
  // 3) ent_emb GEMM (13000 tiles -> 1625 blocks of 8 waves)
  k_entemb<<<1625, 256, 0, stream>>>(local_entity, entity_kge, W_ent, b_ent, entemb);
  // 4) entity_state
  k_estate<<<Bq, 256, 0, stream>>>(q2e, entemb, estate);
  // 5) GRU decoder -> latent
  k_gru<<<Bq, 256, 0, stream>>>(qnode, relproj, W_ih_g, W_hh_g, b_ih_g, b_hh_g,
                                w_state, w_query, latent);
  // 6) scores (16000 waves)
  k_scores<<<2000, 256, 0, stream>>>(estate, latent, entemb, scores);
  // 7) propagation: init + 3 ping-pong hops (one batch per block, 8 rows)
  k_prop_init<<<(Bq * Eq + 255) / 256, 256, 0, stream>>>(q2e, s_a, trav);
  k_prop<<<2000, 256, 0, stream>>>(e2e, s_a, s_b, trav);
  k_prop<<<2000, 256, 0, stream>>>(e2e, s_b, s_a, trav);
  k_prop<<<2000, 256, 0, stream>>>(e2e, s_a, s_b, trav);
  // 8) outputs
  k_final<<<Bq, 256, 0, stream>>>(scores, local_entity, trav, out);
  k_loss<<<1, 256, 0, stream>>>(scores, answer_dist, out);
}